// PointPillarScatter_46042049413720
// MI455X (gfx1250) — compile-verified
//
#include <hip/hip_runtime.h>
#include <hip/hip_bf16.h>

// ---------------------------------------------------------------------------
// PointPillarScatter for MI455X (gfx1250).
// Pure data movement (0 FLOPs): roofline = HBM bandwidth. Strategy: inverse
// index map in workspace, then a fully write-coalesced gather that emits the
// entire 219 MB output as 128-bit non-temporal stores. Index rows are staged
// into LDS with gfx1250 async global->LDS loads (ASYNCcnt) when available.
// ---------------------------------------------------------------------------

#define NXv 432
#define NYv 496
#define Cv  64
#define Bv  4

typedef float f4 __attribute__((ext_vector_type(4)));
typedef int   i4 __attribute__((ext_vector_type(4)));

#define AS1 __attribute__((address_space(1)))
#define AS3 __attribute__((address_space(3)))

#if defined(__AMDGCN__) && __has_builtin(__builtin_amdgcn_global_load_async_to_lds_b128)
#define HAVE_ASYNC_LDS 1
#else
#define HAVE_ASYNC_LDS 0
#endif

// ---- Kernel 1: zero the inverse-index map (3.4 MB, b128 stores) -----------
__global__ __launch_bounds__(256) void pps_zero_map(int* __restrict__ map, int n4) {
    int i = blockIdx.x * 256 + threadIdx.x;
    if (i < n4) {
        i4 z = {0, 0, 0, 0};
        ((i4*)map)[i] = z;
    }
}

// ---- Kernel 2: scatter pillar ids into the map (64000 word stores) --------
__global__ __launch_bounds__(256) void pps_scatter_ids(const int* __restrict__ coords,
                                                       int* __restrict__ map, int P) {
    int p = blockIdx.x * 256 + threadIdx.x;
    if (p >= P) return;
    i4 c = ((const i4*)coords)[p];          // [b, z, y, x]
    // global cell = b*(NY*NX) + (z + y*NX + x)
    int g = c.x * (NYv * NXv) + c.y + c.z * NXv + c.w;
    map[g] = p + 1;                         // 0 == empty cell
}

// ---- Kernel 3: gather + transpose, coalesced NT b128 output stream --------
// One workgroup per (b, y) row; loops all 64 channels for that row.
__global__ __launch_bounds__(256) void pps_gather(const float* __restrict__ feat,
                                                  const int* __restrict__ map,
                                                  float* __restrict__ out) {
    __shared__ int sIdx[NXv];
    const int t = threadIdx.x;
    const int b = blockIdx.x / NYv;
    const int y = blockIdx.x - b * NYv;
    const int* mapRow = map + ((size_t)b * NYv + y) * NXv;

#if HAVE_ASYNC_LDS
    // Stage the 432-entry index row into LDS with async b128 loads.
    if (t < (NXv / 4)) {
        __builtin_amdgcn_global_load_async_to_lds_b128(
            (AS1 i4*)(mapRow + t * 4), (AS3 i4*)(&sIdx[t * 4]), 0, 0);
    }
#if __has_builtin(__builtin_amdgcn_s_wait_asynccnt)
    __builtin_amdgcn_s_wait_asynccnt(0);
#else
    asm volatile("s_wait_asynccnt 0" ::: "memory");
#endif
#else
    for (int i = t; i < NXv; i += 256) sIdx[i] = mapRow[i];
#endif
    __syncthreads();

    float* outBase = out + ((size_t)b * Cv * NYv + y) * NXv;   // + c*NY*NX + x
    constexpr int XQ  = NXv / 4;   // 108 float4 per row
    constexpr int TOT = Cv * XQ;   // 6912 float4 per workgroup (exact 27 iters)
    for (int i = t; i < TOT; i += 256) {
        int c = i / XQ;
        int x = (i - c * XQ) * 4;
        int p0 = sIdx[x + 0], p1 = sIdx[x + 1], p2 = sIdx[x + 2], p3 = sIdx[x + 3];
        f4 v;
        v.x = p0 ? feat[(p0 - 1) * Cv + c] : 0.0f;
        v.y = p1 ? feat[(p1 - 1) * Cv + c] : 0.0f;
        v.z = p2 ? feat[(p2 - 1) * Cv + c] : 0.0f;
        v.w = p3 ? feat[(p3 - 1) * Cv + c] : 0.0f;
        __builtin_nontemporal_store(v, (f4*)(outBase + (size_t)c * (NYv * NXv) + x));
    }
}

// ---- Fallback path (workspace too small): zero-fill + direct scatter ------
__global__ __launch_bounds__(256) void pps_zero_out(float* __restrict__ out, int n4) {
    int i = blockIdx.x * 256 + threadIdx.x;
    if (i < n4) {
        f4 z = {0.f, 0.f, 0.f, 0.f};
        __builtin_nontemporal_store(z, (f4*)out + i);
    }
}

__global__ __launch_bounds__(256) void pps_direct_scatter(const float* __restrict__ feat,
                                                          const int* __restrict__ coords,
                                                          float* __restrict__ out, int P) {
    int gid = blockIdx.x * 256 + threadIdx.x;
    int p = gid >> 6;          // pillar
    int c = gid & (Cv - 1);    // channel
    if (p >= P) return;
    i4 q = ((const i4*)coords)[p];  // [b, z, y, x]
    float v = feat[(size_t)p * Cv + c];
    size_t o = ((size_t)q.x * Cv + c) * (size_t)(NYv * NXv)
             + (size_t)q.y + (size_t)q.z * NXv + (size_t)q.w;
    out[o] = v;
}

extern "C" void kernel_launch(void* const* d_in, const int* in_sizes, int n_in,
                              void* d_out, int out_size, void* d_ws, size_t ws_size,
                              hipStream_t stream) {
    const float* feat   = (const float*)d_in[0];
    const int*   coords = (const int*)d_in[1];
    float*       out    = (float*)d_out;
    const int    P      = in_sizes[0] / Cv;

    const size_t mapBytes = (size_t)Bv * NYv * NXv * sizeof(int);
    if (ws_size >= mapBytes) {
        int* map = (int*)d_ws;
        const int n4 = (Bv * NYv * NXv) / 4;   // 214272, exact
        pps_zero_map<<<(n4 + 255) / 256, 256, 0, stream>>>(map, n4);
        pps_scatter_ids<<<(P + 255) / 256, 256, 0, stream>>>(coords, map, P);
        pps_gather<<<Bv * NYv, 256, 0, stream>>>(feat, map, out);
    } else {
        const int n4 = out_size / 4;
        pps_zero_out<<<(n4 + 255) / 256, 256, 0, stream>>>(out, n4);
        pps_direct_scatter<<<(P * Cv + 255) / 256, 256, 0, stream>>>(feat, coords, out, P);
    }
}